// MultiShapeCrossEntropy_28492813041772
// MI455X (gfx1250) — compile-verified
//
#include <hip/hip_runtime.h>

// Problem shape (fixed by the reference)
#define S_ 16
#define B_ 32
#define P_ 50
#define N_ 2048   // power of two -> cheap div/mod

typedef __attribute__((ext_vector_type(2))) float v2f;
typedef __attribute__((ext_vector_type(8))) float v8f;

// ---------------------------------------------------------------------------
// Kernel 1: per-point cross-entropy, streaming gather-reduce.
// 256 blocks x 256 threads = 65536 threads = one thread per point.
// Lanes cover consecutive n -> every row load is a coalesced 128B/wave access.
// Row values cached in registers (fully unrolled) so HBM is read exactly once
// and only P exps per point are needed (two-pass max + sum-exp).
// ---------------------------------------------------------------------------
__global__ __launch_bounds__(256) void ce_partial_kernel(
    const float* __restrict__ logits,      // [S, B, P, N]
    const int*   __restrict__ points_lab,  // [B, N]
    const int*   __restrict__ shape_lab,   // [B]
    float*       __restrict__ partial)     // [256] block partial sums
{
    __shared__ float red[256];

    const int tid = threadIdx.x;
    const int gid = blockIdx.x * 256 + tid;        // point id in [0, B*N)
    const int b   = gid >> 11;                     // / 2048
    const int n   = gid & (N_ - 1);                // % 2048

    const int shp = shape_lab[b];
    const float* __restrict__ row =
        logits + ((size_t)shp * B_ + (size_t)b) * (size_t)(P_ * N_) + (size_t)n;
    const int lab = points_lab[b * N_ + n];

    float v[P_];
    float m  = -3.402823466e38f;
    float vl = 0.0f;

    #pragma unroll
    for (int p = 0; p < P_; ++p) {
        const float x = row[(size_t)p * N_];       // coalesced across lanes
        v[p] = x;
        m  = fmaxf(m, x);
        vl = (p == lab) ? x : vl;                  // cmp + cndmask, no rescan
    }

    float s = 0.0f;
    #pragma unroll
    for (int p = 0; p < P_; ++p) {
        s += __expf(v[p] - m);                     // v_exp_f32, <= 0 exponent
    }

    const float nll = __logf(s) + m - vl;          // -log_softmax[label]

    // Block tree reduction in LDS.
    red[tid] = nll;
    __syncthreads();
    #pragma unroll
    for (int off = 128; off > 0; off >>= 1) {
        if (tid < off) red[tid] += red[tid + off];
        __syncthreads();
    }
    if (tid == 0) partial[blockIdx.x] = red[0];
}

// ---------------------------------------------------------------------------
// Kernel 2: reduce the 256 block partials with V_WMMA_F32_16X16X4_F32.
// A = ones(16x4), B holds 64 partials (layout-irrelevant for a total sum):
//   D[m][n] = sum_k B[k][n]  -> every row of D is the column-sum vector.
// Four accumulating WMMAs fold all 256 values into C; lanes 0..15 of c[0]
// (row M=0) then hold the 16 column sums, finished with a shfl_xor tree.
// One wave, EXEC all ones as WMMA requires.
// ---------------------------------------------------------------------------
__global__ __launch_bounds__(32) void ce_reduce_kernel(
    const float* __restrict__ partial,   // [256]
    float*       __restrict__ out)       // [1]
{
    const int lane = threadIdx.x;        // 0..31, wave32

    v2f ones; ones.x = 1.0f; ones.y = 1.0f;
    v8f c = {};

    #pragma unroll
    for (int i = 0; i < 4; ++i) {
        v2f bm;
        bm.x = partial[i * 64 + lane];
        bm.y = partial[i * 64 + 32 + lane];
        c = __builtin_amdgcn_wmma_f32_16x16x4_f32(
                /*neg_a=*/false, ones,
                /*neg_b=*/false, bm,
                /*c_mod=*/(short)0, c,
                /*reuse_a=*/false, /*reuse_b=*/false);
    }

    // Row M=0 of D lives in c[0], lanes 0..15 = N=0..15. Sum those 16 lanes.
    float colsum = c[0];
    colsum += __shfl_xor(colsum, 8, 32);
    colsum += __shfl_xor(colsum, 4, 32);
    colsum += __shfl_xor(colsum, 2, 32);
    colsum += __shfl_xor(colsum, 1, 32);

    if (lane == 0) {
        out[0] = colsum * (1.0f / ((float)B_ * (float)N_));   // global mean
    }
}

extern "C" void kernel_launch(void* const* d_in, const int* in_sizes, int n_in,
                              void* d_out, int out_size, void* d_ws, size_t ws_size,
                              hipStream_t stream) {
    const float* logits     = (const float*)d_in[0];  // [S,B,P,N] f32
    const int*   points_lab = (const int*)  d_in[1];  // [B,N] i32
    const int*   shape_lab  = (const int*)  d_in[2];  // [B] i32
    float*       out        = (float*)d_out;
    float*       partial    = (float*)d_ws;           // 256 floats of scratch

    ce_partial_kernel<<<256, 256, 0, stream>>>(logits, points_lab, shape_lab, partial);
    ce_reduce_kernel<<<1, 32, 0, stream>>>(partial, out);
}